// DeepSGC_63677185130847
// MI455X (gfx1250) — compile-verified
//
#include <hip/hip_runtime.h>
#include <hip/hip_bf16.h>

// DeepSGC on MI455X (gfx1250):
//  - 6 graph-propagation hops: wave-per-edge gather + global_atomic_add_f32
//    scatter. h (51.2 MB) is L2-resident (192 MB L2), so this is L2-bound.
//  - 3 dense GEMMs via V_WMMA_F32_16X16X4_F32 (full f32 precision; GEMM is
//    ~7.5 GFLOP total, negligible vs SpMM traffic, so no reason for bf16).
//  - GEMM: each wave owns a 32x16 output block (two 16x16 C tiles sharing the
//    same B operand -> half the W traffic, interleaved WMMA destinations to
//    cover the WMMA->WMMA RAW hazard). 100000 % 32 == 0, so no row guards.

#define FEATS 128

typedef __attribute__((ext_vector_type(2))) float v2f;
typedef __attribute__((ext_vector_type(8))) float v8f;

__global__ void zero_kernel(float* __restrict__ p, long long n) {
  long long i = (long long)blockIdx.x * blockDim.x + threadIdx.x;
  long long stride = (long long)gridDim.x * blockDim.x;
  for (; i < n; i += stride) p[i] = 0.0f;
}

__global__ void deg_kernel(const int* __restrict__ dst, float* __restrict__ deg, int E) {
  int e = blockIdx.x * blockDim.x + threadIdx.x;
  if (e < E) atomicAdd(&deg[dst[e]], 1.0f);
}

__global__ void norm_kernel(float* __restrict__ deg, int n) {
  int i = blockIdx.x * blockDim.x + threadIdx.x;
  if (i < n) {
    float d = deg[i];
    deg[i] = rsqrtf(d < 1.0f ? 1.0f : d);
  }
}

// One wave (32 lanes) per edge; each lane moves a float4 (32*4 = 128 feats).
// Pre-scale by norm[src] fused in. Atomic f32 adds resolve in L2.
__global__ void scatter_kernel(const float* __restrict__ h, const float* __restrict__ norm,
                               const int* __restrict__ src, const int* __restrict__ dst,
                               float* __restrict__ acc, int E) {
  int tid = blockIdx.x * blockDim.x + threadIdx.x;
  int e = tid >> 5;
  if (e >= E) return;
  int lane = tid & 31;
  int s = src[e];
  int d = dst[e];
  float ns = norm[s];
  float4 v = ((const float4*)(h + (size_t)s * FEATS))[lane];
  float* o = acc + (size_t)d * FEATS + lane * 4;
  atomicAdd(o + 0, v.x * ns);
  atomicAdd(o + 1, v.y * ns);
  atomicAdd(o + 2, v.z * ns);
  atomicAdd(o + 3, v.w * ns);
}

// hout = acc * norm[node]; also re-zeroes acc for the next hop.
__global__ void postscale_kernel(float* __restrict__ acc, const float* __restrict__ norm,
                                 float* __restrict__ hout, int n) {
  int tid = blockIdx.x * blockDim.x + threadIdx.x;
  int node = tid >> 5;
  if (node >= n) return;
  int lane = tid & 31;
  float nn = norm[node];
  float4* a4 = (float4*)(acc + (size_t)node * FEATS);
  float4 v = a4[lane];
  v.x *= nn; v.y *= nn; v.z *= nn; v.w *= nn;
  ((float4*)(hout + (size_t)node * FEATS))[lane] = v;
  a4[lane] = make_float4(0.0f, 0.0f, 0.0f, 0.0f);
}

// out[M, ncols] = relu?( A[M,128] @ W[128,ncols] + bias )
// One wave computes a 32x16 output block = two 16x16 WMMA tiles that share
// the B operand. A 16x4 f32 layout: lanes 0-15 hold K=0,1 for row M=lane%16;
// lanes 16-31 hold K=2,3. B mirrors with N=lane%16. C/D: VGPR i holds rows
// i (lanes 0-15) and i+8 (lanes 16-31).
// Requires M % 32 == 0 (100000 = 3125*32), so no row guards anywhere.
__global__ void wmma_gemm_kernel(const float* __restrict__ A, const float* __restrict__ W,
                                 const float* __restrict__ bias, float* __restrict__ out,
                                 int ncols, int tilesN, int numWaves, int relu) {
  int wave = blockIdx.x * (blockDim.x >> 5) + (threadIdx.x >> 5);
  if (wave >= numWaves) return;          // wave-uniform: EXEC stays all-ones
  int lane = threadIdx.x & 31;
  int tn = wave % tilesN;
  int tm = wave / tilesN;                // 32-row block index
  int l16 = lane & 15;
  int half = lane >> 4;                  // 0 or 1
  int kb = half << 1;                    // K offset 0 or 2
  int col = tn * 16 + l16;
  bool colok = col < ncols;
  int colc = colok ? col : 0;
  const float* Wc = W + colc;

  const float* Arow0 = A + (size_t)(tm * 32 + l16) * FEATS;
  const float* Arow1 = Arow0 + (size_t)16 * FEATS;

  float bv = colok ? bias[colc] : 0.0f;
  v8f c0, c1;
#pragma unroll
  for (int i = 0; i < 8; ++i) { c0[i] = bv; c1[i] = bv; }

#pragma unroll 4
  for (int kk = 0; kk < FEATS; kk += 4) {
    v2f a0 = *(const v2f*)(Arow0 + kk + kb);   // 8B-aligned contiguous pair
    v2f a1 = *(const v2f*)(Arow1 + kk + kb);
    float b0 = Wc[(size_t)(kk + kb) * ncols];
    float b1 = Wc[(size_t)(kk + kb + 1) * ncols];
    v2f b;
    b[0] = colok ? b0 : 0.0f;
    b[1] = colok ? b1 : 0.0f;
    c0 = __builtin_amdgcn_wmma_f32_16x16x4_f32(false, a0, false, b,
                                               (short)0, c0, false, false);
    c1 = __builtin_amdgcn_wmma_f32_16x16x4_f32(false, a1, false, b,
                                               (short)0, c1, false, false);
  }
  if (relu) {
#pragma unroll
    for (int i = 0; i < 8; ++i) {
      c0[i] = fmaxf(c0[i], 0.0f);
      c1[i] = fmaxf(c1[i], 0.0f);
    }
  }
  if (colok) {
    // tile0 rows: tm*32 + half*8 + i ; tile1 rows: +16
    float* o0 = out + (size_t)(tm * 32 + half * 8) * ncols + col;
    float* o1 = o0 + (size_t)16 * ncols;
#pragma unroll
    for (int i = 0; i < 8; ++i) {
      o0[(size_t)i * ncols] = c0[i];
      o1[(size_t)i * ncols] = c1[i];
    }
  }
}

extern "C" void kernel_launch(void* const* d_in, const int* in_sizes, int n_in,
                              void* d_out, int out_size, void* d_ws, size_t ws_size,
                              hipStream_t stream) {
  const float* features = (const float*)d_in[0];
  const int*   src      = (const int*)d_in[1];
  const int*   dst      = (const int*)d_in[2];
  const float* W1 = (const float*)d_in[3];
  const float* b1 = (const float*)d_in[4];
  const float* W2 = (const float*)d_in[5];
  const float* b2 = (const float*)d_in[6];
  const float* W3 = (const float*)d_in[7];
  const float* b3 = (const float*)d_in[8];

  const int nN = in_sizes[0] / FEATS;   // 100000 (divisible by 32)
  const int E  = in_sizes[1];           // 1600000
  const int nclasses = in_sizes[8];     // 40

  // Workspace: norm[N], acc[N*128], bufA[N*128], bufB[N*128] (~154 MB)
  char* ws = (char*)d_ws;
  size_t off = 0;
  auto alloc = [&](size_t bytes) -> float* {
    float* p = (float*)(ws + off);
    off = (off + bytes + 255) & ~(size_t)255;
    return p;
  };
  float* norm = alloc((size_t)nN * sizeof(float));
  float* acc  = alloc((size_t)nN * FEATS * sizeof(float));
  float* bufA = alloc((size_t)nN * FEATS * sizeof(float));
  float* bufB = alloc((size_t)nN * FEATS * sizeof(float));
  (void)ws_size; (void)n_in; (void)out_size;

  const int TB = 256;
  zero_kernel<<<1024, TB, 0, stream>>>(norm, nN);
  zero_kernel<<<4096, TB, 0, stream>>>(acc, (long long)nN * FEATS);
  deg_kernel<<<(E + TB - 1) / TB, TB, 0, stream>>>(dst, norm, E);
  norm_kernel<<<(nN + TB - 1) / TB, TB, 0, stream>>>(norm, nN);

  const long long scatThreads = (long long)E * 32;
  const int scatBlocks = (int)((scatThreads + TB - 1) / TB);
  const long long postThreads = (long long)nN * 32;
  const int postBlocks = (int)((postThreads + TB - 1) / TB);

  auto hop = [&](const float* hin, float* hout) {
    scatter_kernel<<<scatBlocks, TB, 0, stream>>>(hin, norm, src, dst, acc, E);
    postscale_kernel<<<postBlocks, TB, 0, stream>>>(acc, norm, hout, nN);
  };

  auto gemm = [&](const float* A, const float* W, const float* bias, float* out,
                  int ncols, int relu) {
    int tilesM = nN / 32;                 // 3125, exact
    int tilesN = (ncols + 15) / 16;
    int waves = tilesM * tilesN;
    int wavesPerBlock = 8;                // 256 threads
    int blocks = (waves + wavesPerBlock - 1) / wavesPerBlock;
    wmma_gemm_kernel<<<blocks, wavesPerBlock * 32, 0, stream>>>(
        A, W, bias, out, ncols, tilesN, waves, relu);
  };

  // Layer 1 (k=2) + ReLU
  hop(features, bufA);
  hop(bufA, bufA);
  gemm(bufA, W1, b1, bufB, FEATS, 1);
  // Layer 2 (k=2) + ReLU
  hop(bufB, bufA);
  hop(bufA, bufA);
  gemm(bufA, W2, b2, bufB, FEATS, 1);
  // Layer 3 (k=2), no activation -> d_out [N, 40]
  hop(bufB, bufA);
  hop(bufA, bufA);
  gemm(bufA, W3, b3, (float*)d_out, nclasses, 0);
}